// QuantumConvLayer_57123065037205
// MI455X (gfx1250) — compile-verified
//
#include <hip/hip_runtime.h>
#include <math.h>

typedef float v2f __attribute__((ext_vector_type(2)));
typedef float v8f __attribute__((ext_vector_type(8)));

// ---------------------------------------------------------------------------
// Setup: build the fixed 16x16 complex matrix  C[:,S] = U2 * B_S * psi0
// from the 28 circuit weights.  Stored transposed for WMMA-B consumption:
//   ws[      S*16 + p] = Re C[p][S]
//   ws[256 + S*16 + p] = Im C[p][S]
// Qubit q acts on bit (3-q) of the state index (wire 0 = MSB).
// ---------------------------------------------------------------------------
namespace {

__device__ __forceinline__ void gate_rx(float re[16], float im[16], float th, int qi) {
  const float c = cosf(0.5f * th);
  const float s = sinf(0.5f * th);
  const int mask = 8 >> qi;
#pragma unroll
  for (int p = 0; p < 16; ++p) {
    if (!(p & mask)) {
      const int q = p | mask;
      const float r0 = re[p], i0 = im[p], r1 = re[q], i1 = im[q];
      // RX = [[c, -i s],[-i s, c]]
      re[p] = c * r0 + s * i1;
      im[p] = c * i0 - s * r1;
      re[q] = c * r1 + s * i0;
      im[q] = c * i1 - s * r0;
    }
  }
}

__device__ __forceinline__ void gate_rz(float re[16], float im[16], float th, int qi) {
  const float c = cosf(0.5f * th);
  const float s = sinf(0.5f * th);
  const int mask = 8 >> qi;
#pragma unroll
  for (int p = 0; p < 16; ++p) {
    const float sg = (p & mask) ? s : -s;   // exp(+i th/2) on bit=1, exp(-i th/2) on bit=0
    const float r = re[p], q = im[p];
    re[p] = r * c - q * sg;
    im[p] = r * sg + q * c;
  }
}

__device__ __forceinline__ void gate_crz(float re[16], float im[16], float th, int ctrl, int tgt) {
  const float c = cosf(0.5f * th);
  const float s = sinf(0.5f * th);
  const int cm = 8 >> ctrl, tm = 8 >> tgt;
#pragma unroll
  for (int p = 0; p < 16; ++p) {
    if (p & cm) {
      const float sg = (p & tm) ? s : -s;
      const float r = re[p], q = im[p];
      re[p] = r * c - q * sg;
      im[p] = r * sg + q * c;
    }
  }
}

} // namespace

extern "C" __global__ void qconv_setup(const float* __restrict__ w, float* __restrict__ ws) {
  const int S = threadIdx.x;   // column index 0..15
  if (S >= 16) return;
  float re[16], im[16];
#pragma unroll
  for (int p = 0; p < 16; ++p) { re[p] = 0.f; im[p] = 0.f; }
  re[0] = 1.f;

  // Layer 1: RX(w[i]) then RZ(w[4+i]) per qubit
#pragma unroll
  for (int i = 0; i < 4; ++i) {
    gate_rx(re, im, w[i], i);
    gate_rz(re, im, w[4 + i], i);
  }
  // 12 CRZ diagonals: ctrl = 3,2,1,0; tgt ascending != ctrl; weights w[8..19]
  int k = 8;
#pragma unroll
  for (int cd = 0; cd < 4; ++cd) {
    const int ctrl = 3 - cd;
#pragma unroll
    for (int tgt = 0; tgt < 4; ++tgt) {
      if (tgt == ctrl) continue;
      gate_crz(re, im, w[k], ctrl, tgt);
      ++k;
    }
  }
  // B_S: apply K = [[0,-1],[1,0]] on every qubit i with bit i of S set
#pragma unroll
  for (int i = 0; i < 4; ++i) {
    if (S & (1 << i)) {
      const int mask = 8 >> i;
#pragma unroll
      for (int p = 0; p < 16; ++p) {
        if (!(p & mask)) {
          const int q = p | mask;
          const float r0 = re[p], i0 = im[p];
          re[p] = -re[q]; im[p] = -im[q];   // K|0> comp gets -a1
          re[q] =  r0;    im[q] =  i0;      // K|1> comp gets  a0
        }
      }
    }
  }
  // U2: RX(w[20+i]) then RZ(w[24+i]) per qubit
#pragma unroll
  for (int i = 0; i < 4; ++i) {
    gate_rx(re, im, w[20 + i], i);
    gate_rz(re, im, w[24 + i], i);
  }
#pragma unroll
  for (int p = 0; p < 16; ++p) {
    ws[S * 16 + p]       = re[p];
    ws[256 + S * 16 + p] = im[p];
  }
}

// ---------------------------------------------------------------------------
// Main: one wave processes 16 patches via fp32 WMMA.
//   A tile : 16 patches x 16 RY-monomials (built in registers)
//   B tile : Cr^T / Ci^T (wave-uniform, from ws)
//   D tile : S_re / S_im -> probs = S_re^2 + S_im^2 -> signed Z-sums via LDS
// ---------------------------------------------------------------------------
extern "C" __global__ __launch_bounds__(256) void qconv_main(
    const float* __restrict__ x, const float* __restrict__ Bmat,
    float* __restrict__ out) {
  __shared__ float lds[8 * 256];

  const int tid  = threadIdx.x;
  const int lane = tid & 31;
  const int wave = tid >> 5;
  const int hb   = lane >> 4;          // half-wave select (K-offset 0 or 2)
  const int M    = lane & 15;          // A row (patch) / B col (state)
  const int base = (blockIdx.x * 8 + wave) * 16;
  const int n    = base + M;           // global patch id, 0..262143
  const int j = n & 63, i = (n >> 6) & 63, b = n >> 12;

  // 2x2 patch from x[b,0,2i:2i+2, 2j:2j+2]; order (0,0),(0,1),(1,0),(1,1)
  const float* img = x + (((size_t)b) << 14) + (size_t)(2 * i) * 128 + (size_t)(2 * j);
  const float2 top = *(const float2*)(img);
  const float2 bot = *(const float2*)(img + 128);
  const float v0 = top.x, v1 = top.y, v2 = bot.x, v3 = bot.y;

  const float ssq  = v0 * v0 + v1 * v1 + v2 * v2 + v3 * v3;
  const bool  zero = (v0 == 0.f) && (v1 == 0.f) && (v2 == 0.f) && (v3 == 0.f);
  const float inv  = zero ? 1.f : (1.f / sqrtf(ssq));
  const float t0 = zero ? 1.f : v0 * inv;
  const float t1 = zero ? 0.f : v1 * inv;
  const float t2 = zero ? 0.f : v2 * inv;
  const float t3 = zero ? 0.f : v3 * inv;

  const float c0 = cosf(0.5f * t0), s0 = sinf(0.5f * t0);
  const float c1 = cosf(0.5f * t1), s1 = sinf(0.5f * t1);
  const float c2 = cosf(0.5f * t2), s2 = sinf(0.5f * t2);
  const float c3 = cosf(0.5f * t3), s3 = sinf(0.5f * t3);

  // m[S] = prod_i (bit_i(S) ? s_i : c_i)
  const float p01[4] = { c0 * c1, s0 * c1, c0 * s1, s0 * s1 };
  const float p23[4] = { c2 * c3, s2 * c3, c2 * s3, s2 * s3 };
  float m[16];
#pragma unroll
  for (int S = 0; S < 16; ++S) m[S] = p01[S & 3] * p23[S >> 2];

  // A fragments (f32 16x4): lane holds K = 4f + 2*hb + {0,1} of row M
  v2f a[4];
#pragma unroll
  for (int f = 0; f < 4; ++f) {
    a[f].x = hb ? m[4 * f + 2] : m[4 * f + 0];
    a[f].y = hb ? m[4 * f + 3] : m[4 * f + 1];
  }

  // B fragments (f32 4x16): lane holds rows K = 4f + 2*hb + {0,1}, col M
  v2f bre[4], bim[4];
#pragma unroll
  for (int f = 0; f < 4; ++f) {
    const int kb = 4 * f + 2 * hb;
    bre[f].x = Bmat[kb * 16 + M];
    bre[f].y = Bmat[kb * 16 + 16 + M];
    bim[f].x = Bmat[256 + kb * 16 + M];
    bim[f].y = Bmat[256 + kb * 16 + 16 + M];
  }

  // S_re = M_tile @ Cr^T ; S_im = M_tile @ Ci^T   (K=16 via 4 chained WMMAs each)
  v8f sre = {0.f, 0.f, 0.f, 0.f, 0.f, 0.f, 0.f, 0.f};
  v8f sim = {0.f, 0.f, 0.f, 0.f, 0.f, 0.f, 0.f, 0.f};
#pragma unroll
  for (int f = 0; f < 4; ++f) {
    sre = __builtin_amdgcn_wmma_f32_16x16x4_f32(false, a[f], false, bre[f],
                                                (short)0, sre, false, false);
    sim = __builtin_amdgcn_wmma_f32_16x16x4_f32(false, a[f], false, bim[f],
                                                (short)0, sim, false, false);
  }
  const v8f P = sre * sre + sim * sim;   // probabilities, D-tile layout

  // Stage 16x16 prob tile: row = patch (v + 8*hb), col = state (M)
  float* lp = &lds[wave * 256];
#pragma unroll
  for (int v = 0; v < 8; ++v) lp[(v + 8 * hb) * 16 + M] = P[v];
  asm volatile("s_wait_dscnt 0x0" ::: "memory");   // per-wave LDS RAW fence

  // Z expectations: out[q] = sum_p (1 - 2*bit_{3-q}(p)) * prob[p]
  if (lane < 16) {
    const float* row = &lds[wave * 256 + lane * 16];
    float q0 = 0.f, q1 = 0.f, q2 = 0.f, q3 = 0.f;
#pragma unroll
    for (int p = 0; p < 16; ++p) {
      const float pv = row[p];
      q0 += (p & 8) ? -pv : pv;
      q1 += (p & 4) ? -pv : pv;
      q2 += (p & 2) ? -pv : pv;
      q3 += (p & 1) ? -pv : pv;
    }
    float* o = out + (((size_t)b) << 14) + (size_t)i * 64 + (size_t)j;
    o[0]     = q0;
    o[4096]  = q1;
    o[8192]  = q2;
    o[12288] = q3;
  }
}

// ---------------------------------------------------------------------------
extern "C" void kernel_launch(void* const* d_in, const int* in_sizes, int n_in,
                              void* d_out, int out_size, void* d_ws, size_t ws_size,
                              hipStream_t stream) {
  const float* x = (const float*)d_in[0];   // (64,1,128,128) fp32
  const float* w = (const float*)d_in[1];   // (28,) fp32
  float* out = (float*)d_out;               // (64,4,64,64) fp32
  float* ws  = (float*)d_ws;                // 2 KB: Cr^T | Ci^T

  qconv_setup<<<1, 16, 0, stream>>>(w, ws);
  // 262144 patches / (8 waves * 16 patches) = 2048 blocks, exact cover
  qconv_main<<<2048, 256, 0, stream>>>(x, ws, out);
}